// wide_deep_17970143167333
// MI455X (gfx1250) — compile-verified
//
#include <hip/hip_runtime.h>
#include <hip/hip_bf16.h>
#include <math.h>

// Model dims
#define Vv   50000
#define Ee   300
#define HH   200     // Hh (per-direction hidden)
#define Bb   256
#define Ss   64
#define EPSf 1e-6f
#define CHP  208     // Hh padded to 13 WMMA tiles of 16
#define KK   (HH + Ee)   // combined K = 500
#define KP2  (KK / 2)    // 250 float2 K-pairs

#define NWORDS ((Vv + 31) / 32)   // 1563 bitset words

typedef __attribute__((ext_vector_type(2))) float v2f;
typedef __attribute__((ext_vector_type(8))) float v8f;

// Workspace layout (in floats).
#define JAC_OFF 0
#define HB_OFF  (JAC_OFF + Bb)                     // 4 dirs * 2 bufs * 256*200 (packed)
#define CB_OFF  (HB_OFF + 4 * 2 * Bb * HH)         // 4 dirs * 256*208
#define MD_OFF  (CB_OFF + 4 * Bb * CHP)            // 4 dirs * 256*64
#define WP_OFF  (MD_OFF + 4 * Bb * Ss)             // packed weights: 4*4*250*208 float2
#define WP_ELEMS (4 * 4 * KP2 * CHP)               // float2 count = 832000

// Packed h-buffer addressing: element (b, ch) lives at [(ch>>1)*512 + b*2 + (ch&1)].
__device__ __forceinline__ int hidx(int b, int ch) {
  return (ch >> 1) * (2 * Bb) + b * 2 + (ch & 1);
}

__device__ __forceinline__ float sigf(float x) { return 1.0f / (1.0f + expf(-x)); }

#define WMMA4(A, B0, B1, B2, B3)                                               \
  do {                                                                         \
    acc0 = __builtin_amdgcn_wmma_f32_16x16x4_f32(false, (A), false, (B0),      \
                                                 (short)0, acc0, false, false);\
    acc1 = __builtin_amdgcn_wmma_f32_16x16x4_f32(false, (A), false, (B1),      \
                                                 (short)0, acc1, false, false);\
    acc2 = __builtin_amdgcn_wmma_f32_16x16x4_f32(false, (A), false, (B2),      \
                                                 (short)0, acc2, false, false);\
    acc3 = __builtin_amdgcn_wmma_f32_16x16x4_f32(false, (A), false, (B3),      \
                                                 (short)0, acc3, false, false);\
  } while (0)

// ---------------------------------------------------------------------------
// Kernel 0: pack [Whh | Wih] per (dir, gate) into WMMA-fragment-native layout:
//   P[d][g][kp][cch] = float2( Wcat[cch][2kp], Wcat[cch][2kp+1] ), cch padded
//   to 208 with zeros.  B-fragment loads become contiguous 128B per half-wave.
// ---------------------------------------------------------------------------
__global__ void pack_weights_kernel(
    const float* __restrict__ Wih0, const float* __restrict__ Whh0,
    const float* __restrict__ Wih1, const float* __restrict__ Whh1,
    const float* __restrict__ Wih2, const float* __restrict__ Whh2,
    const float* __restrict__ Wih3, const float* __restrict__ Whh3,
    float* __restrict__ wp) {
  int idx = blockIdx.x * blockDim.x + threadIdx.x;
  if (idx >= WP_ELEMS) return;
  int cch = idx % CHP;
  int t   = idx / CHP;
  int kp  = t % KP2;
  int g   = (t / KP2) & 3;
  int d   = t / (KP2 * 4);
  const float* Wih = (d == 0) ? Wih0 : (d == 1) ? Wih1 : (d == 2) ? Wih2 : Wih3;
  const float* Whh = (d == 0) ? Whh0 : (d == 1) ? Whh1 : (d == 2) ? Whh2 : Whh3;
  float x = 0.0f, y = 0.0f;
  if (cch < HH) {
    int row = g * HH + cch;
    int k0  = 2 * kp;          // even, so k0 and k0+1 are on the same side of 200
    if (k0 < HH) {
      x = Whh[(size_t)row * HH + k0];
      y = Whh[(size_t)row * HH + k0 + 1];
    } else {
      x = Wih[(size_t)row * Ee + (k0 - HH)];
      y = Wih[(size_t)row * Ee + (k0 - HH) + 1];
    }
  }
  wp[(size_t)idx * 2]     = x;
  wp[(size_t)idx * 2 + 1] = y;
}

// ---------------------------------------------------------------------------
// Kernel 1: per-row jaccard over the 50000-token vocab via LDS bitsets.
// ---------------------------------------------------------------------------
__global__ void jaccard_kernel(const int* __restrict__ t1, const int* __restrict__ t2,
                               float* __restrict__ jac) {
  __shared__ unsigned bits[2 * NWORDS];
  __shared__ int r1[256], r2[256];
  const int b = blockIdx.x, tid = threadIdx.x;
  for (int i = tid; i < 2 * NWORDS; i += blockDim.x) bits[i] = 0u;
  __syncthreads();
  if (tid < Ss) {
    int a = t1[b * Ss + tid];
    int c = t2[b * Ss + tid];
    atomicOr(&bits[a >> 5], 1u << (a & 31));
    atomicOr(&bits[NWORDS + (c >> 5)], 1u << (c & 31));
  }
  __syncthreads();
  int inter = 0, uni = 0;
  for (int i = tid; i < NWORDS; i += blockDim.x) {
    unsigned x = bits[i], y = bits[NWORDS + i];
    inter += __popc(x & y);
    uni   += __popc(x | y);
  }
  r1[tid] = inter; r2[tid] = uni;
  __syncthreads();
  for (int off = 128; off > 0; off >>= 1) {
    if (tid < off) { r1[tid] += r1[tid + off]; r2[tid] += r2[tid + off]; }
    __syncthreads();
  }
  if (tid == 0) jac[b] = (float)r1[0] / (float)r2[0];
}

// ---------------------------------------------------------------------------
// Kernel 2: one workgroup per LSTM direction (d=0:l1f, 1:l1b, 2:l2f, 3:l2b).
// gates = [h_prev | x_s] @ [Whh | Wih]^T + bias via fp32 WMMA 16x16x4.
// Fragment loads are software-pipelined (double-buffered registers) so the
// 4-WMMA group of iteration k overlaps the 5 loads of iteration k+1.
// ---------------------------------------------------------------------------
__global__ void __launch_bounds__(1024)
lstm_kernel(const int* __restrict__ text1, const int* __restrict__ text2,
            const float* __restrict__ emb,
            const float* __restrict__ bs0, const float* __restrict__ bs1,
            const float* __restrict__ bs2, const float* __restrict__ bs3,
            float* __restrict__ ws) {
  const int d = blockIdx.x;
  const float* bs = (d == 0) ? bs0 : (d == 1) ? bs1 : (d == 2) ? bs2 : bs3;
  const int* text = (d < 2) ? text1 : text2;
  const bool rev  = (d & 1) != 0;

  float* hb0 = ws + HB_OFF + (size_t)d * 2 * Bb * HH;   // packed layout
  float* hb1 = hb0 + Bb * HH;
  float* cb  = ws + CB_OFF + (size_t)d * Bb * CHP;
  float* md  = ws + MD_OFF + (size_t)d * Bb * Ss;
  const v2f* w0;
  const v2f* w1;
  const v2f* w2;
  const v2f* w3;
  {
    const v2f* wp = (const v2f*)(ws + WP_OFF) + (size_t)(d * 4) * KP2 * CHP;
    w0 = wp;
    w1 = wp + (size_t)1 * KP2 * CHP;
    w2 = wp + (size_t)2 * KP2 * CHP;
    w3 = wp + (size_t)3 * KP2 * CHP;
  }

  const int tid = threadIdx.x;
  for (int i = tid; i < Bb * HH; i += blockDim.x) hb0[i] = 0.0f;
  for (int i = tid; i < Bb * CHP; i += blockDim.x) cb[i] = 0.0f;
  __threadfence_block();
  __syncthreads();

  __shared__ int tokens[Bb];
  const int wave = tid >> 5;
  const int lane = tid & 31;
  const int half = lane >> 4;
  const int l16  = lane & 15;

  float* hcur = hb0;
  float* hnxt = hb1;

  for (int si = 0; si < Ss; ++si) {
    const int st = rev ? (Ss - 1 - si) : si;
    if (tid < Bb) tokens[tid] = text[tid * Ss + st];
    __syncthreads();

    // 16 M-tiles x 13 channel-tiles = 208 units across 32 waves
    for (int u = wave; u < 16 * 13; u += 32) {
      const int mt = u & 15, ct = u >> 4;
      const int r0   = mt << 4;
      const int brow = r0 + l16;
      const int cch  = (ct << 4) + l16;
      const bool cv  = (cch < HH);
      const int tok  = tokens[brow];
      const float* embrow = emb + (size_t)tok * Ee;
      const v2f* hp = (const v2f*)hcur;

      v8f acc0, acc1, acc2, acc3;
      #pragma unroll
      for (int v = 0; v < 8; ++v) { acc0[v] = 0.0f; acc1[v] = 0.0f; acc2[v] = 0.0f; acc3[v] = 0.0f; }

      // ---- software-pipelined K loop: stage k's loads before k-1's WMMAs ----
      // Prologue: loads for k = 0 of part 1 (recurrent term).
      int kp = half;
      v2f a  = hp[kp * Bb + brow];
      v2f b0 = w0[kp * CHP + cch];
      v2f b1 = w1[kp * CHP + cch];
      v2f b2 = w2[kp * CHP + cch];
      v2f b3 = w3[kp * CHP + cch];

      // Part 1 steady state: k = 1..49 (A from packed h_prev).
      for (int k = 1; k < HH / 4; ++k) {
        const int kpn = 2 * k + half;
        v2f an = hp[kpn * Bb + brow];
        v2f c0 = w0[kpn * CHP + cch];
        v2f c1 = w1[kpn * CHP + cch];
        v2f c2 = w2[kpn * CHP + cch];
        v2f c3 = w3[kpn * CHP + cch];
        WMMA4(a, b0, b1, b2, b3);
        a = an; b0 = c0; b1 = c1; b2 = c2; b3 = c3;
      }
      // Boundary: issue part-2 k=0 loads (A from embedding row), WMMA part-1 k=49.
      {
        const int kpn = (HH / 2) + half;
        v2f an;
        an.x = embrow[2 * half];
        an.y = embrow[2 * half + 1];
        v2f c0 = w0[kpn * CHP + cch];
        v2f c1 = w1[kpn * CHP + cch];
        v2f c2 = w2[kpn * CHP + cch];
        v2f c3 = w3[kpn * CHP + cch];
        WMMA4(a, b0, b1, b2, b3);
        a = an; b0 = c0; b1 = c1; b2 = c2; b3 = c3;
      }
      // Part 2 steady state: k = 1..74 (A from embedding row gather).
      for (int k = 1; k < Ee / 4; ++k) {
        const int kkn = 4 * k + 2 * half;
        const int kpn = (HH / 2) + 2 * k + half;
        v2f an;
        an.x = embrow[kkn];
        an.y = embrow[kkn + 1];
        v2f c0 = w0[kpn * CHP + cch];
        v2f c1 = w1[kpn * CHP + cch];
        v2f c2 = w2[kpn * CHP + cch];
        v2f c3 = w3[kpn * CHP + cch];
        WMMA4(a, b0, b1, b2, b3);
        a = an; b0 = c0; b1 = c1; b2 = c2; b3 = c3;
      }
      // Epilogue: final WMMA group (k = 74 of part 2).
      WMMA4(a, b0, b1, b2, b3);

      if (cv) {
        const float bi = bs[cch];
        const float bf = bs[HH + cch];
        const float bg = bs[2 * HH + cch];
        const float bo = bs[3 * HH + cch];
        #pragma unroll
        for (int v = 0; v < 8; ++v) {
          const int m = r0 + v + 8 * half;       // D-tile row -> batch index
          float gi = acc0[v] + bi;
          float gf = acc1[v] + bf;
          float gg = acc2[v] + bg;
          float go = acc3[v] + bo;
          float cold = cb[m * CHP + cch];
          float cnew = sigf(gf) * cold + sigf(gi) * tanhf(gg);
          cb[m * CHP + cch] = cnew;
          hnxt[hidx(m, cch)] = sigf(go) * tanhf(cnew);
        }
      }
    }
    __threadfence_block();
    __syncthreads();

    // per-(b, s) max over this direction's 200 output channels
    if (tid < Bb) {
      float mx = -INFINITY;
      for (int ch = 0; ch < HH; ++ch) mx = fmaxf(mx, hnxt[hidx(tid, ch)]);
      md[tid * Ss + st] = mx;
    }
    float* t = hcur; hcur = hnxt; hnxt = t;
  }
  // After 64 steps hcur == hb0: final hidden state lands in buffer 0 (packed).
}

// ---------------------------------------------------------------------------
// Kernel 3: head. One workgroup per batch row: wide features, 5-layer MLP,
// final 205->2 projection, log-softmax.
// ---------------------------------------------------------------------------
__global__ void head_kernel(float* __restrict__ ws,
    const float* __restrict__ w1, const float* __restrict__ b1,
    const float* __restrict__ w2, const float* __restrict__ b2,
    const float* __restrict__ w3, const float* __restrict__ b3,
    const float* __restrict__ w4, const float* __restrict__ b4,
    const float* __restrict__ w5, const float* __restrict__ b5,
    const float* __restrict__ wm, const float* __restrict__ bm,
    float* __restrict__ out) {
  const int b = blockIdx.x, tid = threadIdx.x;
  __shared__ float vec[4 * HH];   // deep = [s1 | s2] (800), reused as MLP buffer
  __shared__ float alt[HH];
  __shared__ float red[256];
  __shared__ float m1s[Ss], m2s[Ss];
  __shared__ float wide[5];

  const float* hT0 = ws + HB_OFF + 0 * 2 * Bb * HH;   // final h in buffer 0, packed
  const float* hT1 = ws + HB_OFF + 1 * 2 * Bb * HH;
  const float* hT2 = ws + HB_OFF + 2 * 2 * Bb * HH;
  const float* hT3 = ws + HB_OFF + 3 * 2 * Bb * HH;
  if (tid < HH) {
    vec[tid]          = hT0[hidx(b, tid)];
    vec[HH + tid]     = hT1[hidx(b, tid)];
    vec[2 * HH + tid] = hT2[hidx(b, tid)];
    vec[3 * HH + tid] = hT3[hidx(b, tid)];
  }
  if (tid < Ss) {
    const float* md = ws + MD_OFF;
    m1s[tid] = fmaxf(md[0 * Bb * Ss + b * Ss + tid], md[1 * Bb * Ss + b * Ss + tid]);
    m2s[tid] = fmaxf(md[2 * Bb * Ss + b * Ss + tid], md[3 * Bb * Ss + b * Ss + tid]);
  }
  __syncthreads();

  // dot / dist over 400 dims of s1 vs s2
  float pd = 0.0f, pq = 0.0f;
  for (int j = tid; j < 2 * HH; j += blockDim.x) {
    float x = vec[j], y = vec[2 * HH + j];
    pd += x * y;
    float df = x - y + EPSf;
    pq += df * df;
  }
  red[tid] = pd; __syncthreads();
  for (int o = 128; o > 0; o >>= 1) { if (tid < o) red[tid] += red[tid + o]; __syncthreads(); }
  if (tid == 0) wide[0] = red[0];
  __syncthreads();
  red[tid] = pq; __syncthreads();
  for (int o = 128; o > 0; o >>= 1) { if (tid < o) red[tid] += red[tid + o]; __syncthreads(); }
  if (tid == 0) wide[1] = sqrtf(red[0]);
  __syncthreads();

  // mdot / mdist over the 64 per-step maxima
  float pm = 0.0f, pmq = 0.0f;
  if (tid < Ss) {
    float x = m1s[tid], y = m2s[tid];
    pm = x * y;
    float df = x - y + EPSf;
    pmq = df * df;
  }
  red[tid] = pm; __syncthreads();
  for (int o = 128; o > 0; o >>= 1) { if (tid < o) red[tid] += red[tid + o]; __syncthreads(); }
  if (tid == 0) wide[2] = red[0];
  __syncthreads();
  red[tid] = pmq; __syncthreads();
  for (int o = 128; o > 0; o >>= 1) { if (tid < o) red[tid] += red[tid + o]; __syncthreads(); }
  if (tid == 0) { wide[3] = sqrtf(red[0]); wide[4] = ws[JAC_OFF + b]; }
  __syncthreads();

  // 5-layer MLP: 800->200 (relu), 200->200 (relu) x3, then 200->200 linear
  const float* Wl[5] = {w1, w2, w3, w4, w5};
  const float* Bl[5] = {b1, b2, b3, b4, b5};
  const int din[5]   = {4 * HH, HH, HH, HH, HH};
  float* cur = vec;
  float* nxt = alt;
  for (int L = 0; L < 5; ++L) {
    float a = 0.0f;
    if (tid < HH) {
      a = Bl[L][tid];
      const float* wr = Wl[L] + (size_t)tid * din[L];
      for (int k = 0; k < din[L]; ++k) a += cur[k] * wr[k];
      if (L < 4) a = fmaxf(a, 0.0f);
    }
    __syncthreads();
    if (tid < HH) nxt[tid] = a;
    __syncthreads();
    float* t = cur; cur = nxt; nxt = t;
  }

  if (tid == 0) {
    float o0 = bm[0], o1 = bm[1];
    for (int k = 0; k < HH; ++k) {
      o0 += cur[k] * wm[k];
      o1 += cur[k] * wm[205 + k];
    }
    for (int q = 0; q < 5; ++q) {
      o0 += wide[q] * wm[200 + q];
      o1 += wide[q] * wm[205 + 200 + q];
    }
    float mx  = fmaxf(o0, o1);
    float lse = mx + logf(expf(o0 - mx) + expf(o1 - mx));
    out[b * 2 + 0] = o0 - lse;
    out[b * 2 + 1] = o1 - lse;
  }
}

// ---------------------------------------------------------------------------
extern "C" void kernel_launch(void* const* d_in, const int* in_sizes, int n_in,
                              void* d_out, int out_size, void* d_ws, size_t ws_size,
                              hipStream_t stream) {
  (void)in_sizes; (void)n_in; (void)out_size; (void)ws_size;
  const int*   text1   = (const int*)  d_in[0];
  const int*   text2   = (const int*)  d_in[1];
  const float* emb     = (const float*)d_in[2];
  const float* l1f_Wih = (const float*)d_in[3];
  const float* l1f_Whh = (const float*)d_in[4];
  const float* l1f_b   = (const float*)d_in[5];
  const float* l1b_Wih = (const float*)d_in[6];
  const float* l1b_Whh = (const float*)d_in[7];
  const float* l1b_b   = (const float*)d_in[8];
  const float* l2f_Wih = (const float*)d_in[9];
  const float* l2f_Whh = (const float*)d_in[10];
  const float* l2f_b   = (const float*)d_in[11];
  const float* l2b_Wih = (const float*)d_in[12];
  const float* l2b_Whh = (const float*)d_in[13];
  const float* l2b_b   = (const float*)d_in[14];
  const float* w1 = (const float*)d_in[15];
  const float* b1 = (const float*)d_in[16];
  const float* w2 = (const float*)d_in[17];
  const float* b2 = (const float*)d_in[18];
  const float* w3 = (const float*)d_in[19];
  const float* b3 = (const float*)d_in[20];
  const float* w4 = (const float*)d_in[21];
  const float* b4 = (const float*)d_in[22];
  const float* w5 = (const float*)d_in[23];
  const float* b5 = (const float*)d_in[24];
  const float* wm = (const float*)d_in[25];
  const float* bm = (const float*)d_in[26];

  float* ws  = (float*)d_ws;
  float* out = (float*)d_out;

  pack_weights_kernel<<<(WP_ELEMS + 255) / 256, 256, 0, stream>>>(
      l1f_Wih, l1f_Whh, l1b_Wih, l1b_Whh,
      l2f_Wih, l2f_Whh, l2b_Wih, l2b_Whh, ws + WP_OFF);
  jaccard_kernel<<<Bb, 256, 0, stream>>>(text1, text2, ws + JAC_OFF);
  lstm_kernel<<<4, 1024, 0, stream>>>(text1, text2, emb,
      l1f_b, l1b_b, l2f_b, l2b_b, ws);
  head_kernel<<<Bb, 256, 0, stream>>>(ws, w1, b1, w2, b2, w3, b3, w4, b4,
                                      w5, b5, wm, bm, out);
}